// NAM_49314814493074
// MI455X (gfx1250) — compile-verified
//
#include <hip/hip_runtime.h>

// CDNA5 / gfx1250 fused per-feature MLP:  out = sum_f w[b,f],  w = MLP_f(x[b,f])
// - Layer-2 as D' = blockdiag(W2^T) x H1^T on V_WMMA_F32_16X16X4_F32 (layer-3
//   dot is a within-lane reduction over the 8 accumulator VGPRs).
// - Weights double-buffered in LDS via GLOBAL_LOAD_ASYNC_TO_LDS_B128,
//   pipelined one chunk ahead of compute (ASYNCcnt).
// - w tile written back with GLOBAL_STORE_ASYNC_FROM_LDS_B128.

typedef float v2f __attribute__((ext_vector_type(2)));
typedef float v4f __attribute__((ext_vector_type(4)));
typedef float v8f __attribute__((ext_vector_type(8)));
typedef int   v4i __attribute__((ext_vector_type(4)));

constexpr int kB = 32768;
constexpr int kF = 256;
constexpr int kH = 8;
constexpr int kChunk = 32;                           // features per LDS chunk
constexpr int kNChunks = kF / kChunk;                // 8
constexpr int kWaves = 4;                            // 128 threads / wave32
constexpr int kRowsPerWave = 16;                     // WMMA N dimension (batch)
constexpr int kRowsPerBlock = kWaves * kRowsPerWave; // 64
constexpr int kThreads = kWaves * 32;                // 128
constexpr int kWPad = kChunk + 4;                    // 36: bank-spread row stride

#if defined(__HIP_DEVICE_COMPILE__) && \
    __has_builtin(__builtin_amdgcn_global_load_async_to_lds_b128)
#define ASYNC_LD 1
#else
#define ASYNC_LD 0
#endif
#if defined(__HIP_DEVICE_COMPILE__) && \
    __has_builtin(__builtin_amdgcn_global_store_async_from_lds_b128)
#define ASYNC_ST 1
#else
#define ASYNC_ST 0
#endif
#if ASYNC_LD || ASYNC_ST
typedef __attribute__((address_space(1))) v4i GV4;   // global int4
typedef __attribute__((address_space(3))) v4i LV4;   // LDS int4
#endif

static __device__ inline void wait_async0() {
#if ASYNC_LD || ASYNC_ST
#if __has_builtin(__builtin_amdgcn_s_wait_asynccnt)
  __builtin_amdgcn_s_wait_asynccnt(0);
#else
  asm volatile("s_wait_asynccnt 0x0" ::: "memory");
#endif
#endif
}

__global__ __launch_bounds__(kThreads, 1) void fused_feature_mlp(
    const float* __restrict__ x,
    const float* __restrict__ W1, const float* __restrict__ b1,
    const float* __restrict__ W2, const float* __restrict__ b2,
    const float* __restrict__ W3,
    float* __restrict__ out_sum, float* __restrict__ out_w)
{
  __shared__ __align__(16) float sW1[2][kChunk * kH];
  __shared__ __align__(16) float sB1[2][kChunk * kH];
  __shared__ __align__(16) float sB2[2][kChunk * kH];
  __shared__ __align__(16) float sW3[2][kChunk * kH];
  __shared__ __align__(16) float sW2[2][kChunk * kH * kH];
  __shared__ __align__(16) float wbuf[kWaves][kRowsPerWave][kWPad];

  const int tid  = threadIdx.x;
  const int lane = tid & 31;
  const int wave = tid >> 5;
  const int m    = lane & 15;        // batch row within tile (B/C/D column N)
  const int hf   = lane >> 4;        // lane-half selector for VGPR striping
  const int fsel = (lane >> 3) & 1;  // A-matrix row mo >= 8 -> second feature
  const int hcol = lane & 7;         // mo & 7: output hidden unit of A row

  const int waveRowBase = blockIdx.x * kRowsPerBlock + wave * kRowsPerWave;
  const float* xrow = x + (size_t)(waveRowBase + m) * kF;

  // stage one chunk of weights into LDS buffer `buf`
  auto stage = [&](int buf, int cbase) {
#if ASYNC_LD
    if (tid < kChunk * kH / 4) {  // 256 floats per small array = 64 float4
      const int t4 = tid * 4;
      __builtin_amdgcn_global_load_async_to_lds_b128((GV4*)(W1 + cbase * kH + t4), (LV4*)&sW1[buf][t4], 0, 0);
      __builtin_amdgcn_global_load_async_to_lds_b128((GV4*)(b1 + cbase * kH + t4), (LV4*)&sB1[buf][t4], 0, 0);
      __builtin_amdgcn_global_load_async_to_lds_b128((GV4*)(b2 + cbase * kH + t4), (LV4*)&sB2[buf][t4], 0, 0);
      __builtin_amdgcn_global_load_async_to_lds_b128((GV4*)(W3 + cbase * kH + t4), (LV4*)&sW3[buf][t4], 0, 0);
    }
#pragma unroll
    for (int i = 0; i < 4; ++i) {  // 2048 floats of W2 = 512 float4
      int idx = (tid + i * kThreads) * 4;
      __builtin_amdgcn_global_load_async_to_lds_b128((GV4*)(W2 + cbase * kH * kH + idx), (LV4*)&sW2[buf][idx], 0, 0);
    }
#else
    for (int i = tid; i < kChunk * kH; i += kThreads) {
      sW1[buf][i] = W1[cbase * kH + i];
      sB1[buf][i] = b1[cbase * kH + i];
      sB2[buf][i] = b2[cbase * kH + i];
      sW3[buf][i] = W3[cbase * kH + i];
    }
    for (int i = tid; i < kChunk * kH * kH; i += kThreads)
      sW2[buf][i] = W2[cbase * kH * kH + i];
#endif
  };

  float acc = 0.0f;  // running sum over this lane's feature parity

  stage(0, 0);  // prologue: chunk 0 in flight

  for (int c = 0; c < kNChunks; ++c) {
    const int cb  = c * kChunk;
    const int cur = c & 1;

    wait_async0();    // chunk c weights landed; prior w-tile stores drained
    __syncthreads();  // all waves see staged weights; ping buffer reusable

    if (c + 1 < kNChunks) stage(cur ^ 1, cb + kChunk);  // overlap DMA w/ compute

    // ---- process features two at a time ----
#pragma unroll 2
    for (int p = 0; p < kChunk / 2; ++p) {
      const int fl0 = 2 * p;  // local (in-chunk) feature index of the pair
      v2f xv = *(const v2f*)(xrow + cb + fl0);
      __builtin_prefetch(xrow + cb + fl0 + 2, 0, 3);  // global_prefetch_b8 (near)

      // B fragment = H1^T: row K = 4i + 2v + half, col N = lane&15 (batch)
      v2f bm[4];
#pragma unroll
      for (int i = 0; i < 4; ++i) {
#pragma unroll
        for (int v = 0; v < 2; ++v) {
          int k  = 4 * i + 2 * v + hf;
          int kf = k >> 3;
          int kh = k & 7;
          float xs = kf ? xv.y : xv.x;
          float h1 = xs * sW1[cur][(fl0 + kf) * kH + kh] + sB1[cur][(fl0 + kf) * kH + kh];
          bm[i][v] = h1 > 0.0f ? h1 : 0.0f;
        }
      }
      // A fragment = blockdiag(W2^T): unconditional load + cndmask select
      v2f am[4];
#pragma unroll
      for (int i = 0; i < 4; ++i) {
#pragma unroll
        for (int v = 0; v < 2; ++v) {
          int k  = 4 * i + v + 2 * hf;
          int kf = k >> 3;
          int kh = k & 7;
          float w2v = sW2[cur][(fl0 + fsel) * kH * kH + kh * kH + hcol];
          am[i][v] = (kf == fsel) ? w2v : 0.0f;
        }
      }
      // C: per-VGPR bias b2[f0+hf][j]
      v4f clo = *(const v4f*)&sB2[cur][(fl0 + hf) * kH];
      v4f chi = *(const v4f*)&sB2[cur][(fl0 + hf) * kH + 4];
      v8f d = {clo.x, clo.y, clo.z, clo.w, chi.x, chi.y, chi.z, chi.w};
      d = __builtin_amdgcn_wmma_f32_16x16x4_f32(false, am[0], false, bm[0], (short)0, d, false, false);
      d = __builtin_amdgcn_wmma_f32_16x16x4_f32(false, am[1], false, bm[1], (short)0, d, false, false);
      d = __builtin_amdgcn_wmma_f32_16x16x4_f32(false, am[2], false, bm[2], (short)0, d, false, false);
      d = __builtin_amdgcn_wmma_f32_16x16x4_f32(false, am[3], false, bm[3], (short)0, d, false, false);

      // Layer 3: w = sum_j relu(D'[j]) * W3[f0+hf][j]  -- all within-lane
      v4f w3lo = *(const v4f*)&sW3[cur][(fl0 + hf) * kH];
      v4f w3hi = *(const v4f*)&sW3[cur][(fl0 + hf) * kH + 4];
      float w = 0.0f;
      w = fmaf(fmaxf(d[0], 0.0f), w3lo.x, w);
      w = fmaf(fmaxf(d[1], 0.0f), w3lo.y, w);
      w = fmaf(fmaxf(d[2], 0.0f), w3lo.z, w);
      w = fmaf(fmaxf(d[3], 0.0f), w3lo.w, w);
      w = fmaf(fmaxf(d[4], 0.0f), w3hi.x, w);
      w = fmaf(fmaxf(d[5], 0.0f), w3hi.y, w);
      w = fmaf(fmaxf(d[6], 0.0f), w3hi.z, w);
      w = fmaf(fmaxf(d[7], 0.0f), w3hi.w, w);

      acc += w;
      // every lane owns one (row, feature) result; conflict-free padded stride
      wbuf[wave][m][fl0 + hf] = w;
    }

    // ---- flush this wave's 16x32 w tile (wave-private: no barrier needed) ----
#if ASYNC_ST
    // DS writes must be in LDS before the async engine reads them
#if __has_builtin(__builtin_amdgcn_s_wait_dscnt)
    __builtin_amdgcn_s_wait_dscnt(0);
#else
    asm volatile("s_wait_dscnt 0x0" ::: "memory");
#endif
#pragma unroll
    for (int i = 0; i < 4; ++i) {
      int idx = i * 128 + lane * 4;  // 512 floats / wave, float4 per lane
      int r   = idx >> 5;
      int fc  = idx & 31;
      __builtin_amdgcn_global_store_async_from_lds_b128(
          (GV4*)(out_w + (size_t)(waveRowBase + r) * kF + cb + fc),
          (LV4*)&wbuf[wave][r][fc], 0, 0);
    }
#else
#pragma unroll
    for (int i = 0; i < 4; ++i) {
      int idx = i * 128 + lane * 4;
      int r   = idx >> 5;
      int fc  = idx & 31;
      v4f val = *(const v4f*)&wbuf[wave][r][fc];
      *(v4f*)(out_w + (size_t)(waveRowBase + r) * kF + cb + fc) = val;
    }
#endif
  }

  // ---- per-row output: merge the two feature-parity halves of the wave ----
  acc += __shfl_xor(acc, 16);
  if (lane < 16) out_sum[waveRowBase + lane] = acc;
  // S_ENDPGM's implicit wait-idle drains outstanding async stores.
}

extern "C" void kernel_launch(void* const* d_in, const int* in_sizes, int n_in,
                              void* d_out, int out_size, void* d_ws, size_t ws_size,
                              hipStream_t stream) {
  const float* x  = (const float*)d_in[0];
  const float* W1 = (const float*)d_in[1];
  const float* b1 = (const float*)d_in[2];
  const float* W2 = (const float*)d_in[3];
  const float* b2 = (const float*)d_in[4];
  const float* W3 = (const float*)d_in[5];
  float* out_sum = (float*)d_out;       // [B] first
  float* out_w   = out_sum + kB;        // then [B, F]

  dim3 grid(kB / kRowsPerBlock);        // 512 blocks x 128 threads (4 waves)
  fused_feature_mlp<<<grid, kThreads, 0, stream>>>(x, W1, b1, W2, b2, W3,
                                                   out_sum, out_w);
}